// PerObjectFiLM_20907900797339
// MI455X (gfx1250) — compile-verified
//
#include <hip/hip_runtime.h>

typedef __attribute__((ext_vector_type(2))) float v2f;
typedef __attribute__((ext_vector_type(8))) float v8f;

#define B_   16
#define K_   16
#define CH   128
#define HW_  4096
#define DT   512
#define DO_  128
#define DIN  640
#define DOUT 256

// Kernel 1: film = [task_emb | obj_tokens] @ film_w + film_b
// M = B*K = 256 rows, N = 256 cols, K = 640.
// 4 waves per block; each wave owns one 16x16 output tile and runs the
// K-reduction with V_WMMA_F32_16X16X4_F32. The K loop is split at the concat
// boundary (512, a multiple of the K-step) so the A-operand source is
// wave-uniform per iteration -> no divergent EXEC churn in the hot loop.
__global__ void film_gemm_kernel(const float* __restrict__ task_emb,
                                 const float* __restrict__ obj_tokens,
                                 const float* __restrict__ film_w,
                                 const float* __restrict__ film_b,
                                 float* __restrict__ film_out) {
    const int lane = threadIdx.x & 31;
    const int wave = threadIdx.x >> 5;          // 0..3
    const int l16  = lane & 15;
    const int half = lane >> 4;                 // 0 or 1
    const int m0   = blockIdx.x * 16;           // row tile over (b,k)
    const int n0   = (blockIdx.y * 4 + wave) * 16;  // col tile over 2*C_h

    // A-matrix row this lane supplies: M = l16
    const int row  = m0 + l16;
    const int bidx = row >> 4;   // batch index (rows are b*16 + k)
    const int kidx = row & 15;   // object index

    const float* ta = task_emb   + (size_t)bidx * DT;
    const float* ot = obj_tokens + (size_t)kidx * DO_;

    v8f acc = {};
    // --- columns [0, 512): A comes from task_emb ---
    for (int kk = 0; kk < DT; kk += 4) {
        v2f a, w;
#pragma unroll
        for (int v = 0; v < 2; ++v) {
            const int col = kk + v + 2 * half;       // A 16x4: VGPR v, lanes>=16 -> K+2
            a[v] = ta[col];
            w[v] = film_w[(size_t)col * DOUT + n0 + l16];  // B 4x16: lane = N
        }
        acc = __builtin_amdgcn_wmma_f32_16x16x4_f32(
            false, a, false, w, (short)0, acc, false, false);
    }
    // --- columns [512, 640): A comes from obj_tokens ---
    for (int kk = DT; kk < DIN; kk += 4) {
        v2f a, w;
#pragma unroll
        for (int v = 0; v < 2; ++v) {
            const int col = kk + v + 2 * half;
            a[v] = ot[col - DT];
            w[v] = film_w[(size_t)col * DOUT + n0 + l16];
        }
        acc = __builtin_amdgcn_wmma_f32_16x16x4_f32(
            false, a, false, w, (short)0, acc, false, false);
    }

    const float bias = film_b[n0 + l16];
#pragma unroll
    for (int r = 0; r < 8; ++r) {
        // C/D 16x16 f32 layout: M = r + 8*half, N = l16
        const int M = m0 + r + 8 * half;
        film_out[(size_t)M * DOUT + n0 + l16] = acc[r] + bias;
    }
}

// Kernel 2 (bandwidth kernel): for each batch b and 16-wide HW tile,
//   out[b,c,hw] = (sum_k gamma[b,k,c]*mask[b,k,hw]) * h_base[b,c,hw]
//              +  (sum_k  beta[b,k,c]*mask[b,k,hw])
// 4 waves per block; each wave owns one (b, hw-tile). The 16(k) x 16(hw) mask
// tile lives in the B-operand VGPRs (4 k-step operands) and is reused across
// all 8 channel tiles and both gamma/beta accumulators (16x register reuse).
// The FiLM apply is fused: one h_base read + one out write per element;
// g_map/b_map never touch memory (saves ~134 MB of intermediate traffic).
__global__ void film_apply_kernel(const float* __restrict__ film,     // (B*K, 256) = [gamma|beta]
                                  const float* __restrict__ obj_mask, // (B, K, HW)
                                  const float* __restrict__ h_base,   // (B, CH, HW)
                                  float* __restrict__ out) {          // (B, CH, HW)
    const int lane = threadIdx.x & 31;
    const int wave = threadIdx.x >> 5;                  // 0..3
    const int l16  = lane & 15;
    const int half = lane >> 4;
    const int hw0  = blockIdx.x * 64 + wave * 16;
    const int b    = blockIdx.y;

    // Load the full 16(k) x 16(hw) mask tile into 4 k-step B operands.
    v2f mk[4];
#pragma unroll
    for (int s = 0; s < 4; ++s) {
#pragma unroll
        for (int v = 0; v < 2; ++v) {
            const int k = s * 4 + v + 2 * half;   // B layout: lane=N, VGPR+half pick K
            mk[s][v] = obj_mask[(size_t)(b * K_ + k) * HW_ + hw0 + l16];
        }
    }

    for (int ct = 0; ct < CH / 16; ++ct) {
        const int c0 = ct * 16;
        v8f g = {}, be = {};
#pragma unroll
        for (int s = 0; s < 4; ++s) {
            v2f ag, ab;
#pragma unroll
            for (int v = 0; v < 2; ++v) {
                const int k = s * 4 + v + 2 * half;   // A layout: lane%16 = M (=c), VGPR = K (=k)
                const float* frow = film + (size_t)(b * K_ + k) * DOUT;
                ag[v] = frow[c0 + l16];        // gamma
                ab[v] = frow[CH + c0 + l16];   // beta
            }
            g  = __builtin_amdgcn_wmma_f32_16x16x4_f32(false, ag, false, mk[s],
                                                       (short)0, g,  false, false);
            be = __builtin_amdgcn_wmma_f32_16x16x4_f32(false, ab, false, mk[s],
                                                       (short)0, be, false, false);
        }
#pragma unroll
        for (int r = 0; r < 8; ++r) {
            const int c = c0 + r + 8 * half;          // C/D layout: M = r + 8*half
            const size_t idx = ((size_t)(b * CH + c)) * HW_ + hw0 + l16;
            out[idx] = g[r] * h_base[idx] + be[r];
        }
    }
}

extern "C" void kernel_launch(void* const* d_in, const int* in_sizes, int n_in,
                              void* d_out, int out_size, void* d_ws, size_t ws_size,
                              hipStream_t stream) {
    const float* task_emb   = (const float*)d_in[0];
    const float* h_base     = (const float*)d_in[1];
    const float* obj_mask   = (const float*)d_in[2];
    const float* film_w     = (const float*)d_in[3];
    const float* film_b     = (const float*)d_in[4];
    const float* obj_tokens = (const float*)d_in[5];
    float* out     = (float*)d_out;
    float* film_ws = (float*)d_ws;   // (B*K) x 256 f32 = 256 KB scratch

    // Stage 1: film GEMM. 16 M-tiles x (4 N-tile groups), 4 waves per block.
    film_gemm_kernel<<<dim3((B_ * K_) / 16, DOUT / 64), 128, 0, stream>>>(
        task_emb, obj_tokens, film_w, film_b, film_ws);

    // Stage 2: einsum via WMMA + fused FiLM apply. 4 hw-tiles per block.
    film_apply_kernel<<<dim3(HW_ / 64, B_), 128, 0, stream>>>(
        film_ws, obj_mask, h_base, out);
}